// GatedBlock_85847806312474
// MI455X (gfx1250) — compile-verified
//
#include <hip/hip_runtime.h>
#include <math.h>

typedef __attribute__((ext_vector_type(2))) float v2f;
typedef __attribute__((ext_vector_type(8))) float v8f;

#define B_ 16
#define C_ 128
#define T_ 64
#define F_ 128
#define D_ 2
#define S_ 62

// ---- branchless hardware transcendentals --------------------------------
#if __has_builtin(__builtin_amdgcn_tanhf)
__device__ __forceinline__ float fast_tanh(float x) {
    return __builtin_amdgcn_tanhf(x);          // v_tanh_f32 (gfx1250 TRANS op)
}
#else
__device__ __forceinline__ float fast_tanh(float x) {
    // tanh(x) = 1 - 2/(exp2(2*log2(e)*x) + 1), branchless via v_exp_f32/v_rcp_f32
    const float e = __builtin_amdgcn_exp2f(x * 2.885390081777927e0f); // 2/ln(2)
    return 1.0f - 2.0f * __builtin_amdgcn_rcpf(e + 1.0f);
}
#endif
__device__ __forceinline__ float fast_sigmoid(float x) {
    return fmaf(fast_tanh(0.5f * x), 0.5f, 0.5f);  // sigmoid = 0.5*(1+tanh(x/2))
}

// One block per (b,c) slice. 8 waves/block; wave w owns F-columns [16w,16w+16).
// res = w_res(62x64, pad->64) @ x_slice(64x128) via V_WMMA_F32_16X16X4_F32,
// fused with the tanh*sigmoid gate and output scale/bias epilogue.
__global__ __launch_bounds__(256, 2)
void gated_block_kernel(const float* __restrict__ x,
                        const float* __restrict__ w_tanh,
                        const float* __restrict__ b_tanh,
                        const float* __restrict__ w_sig,
                        const float* __restrict__ b_sig,
                        const float* __restrict__ w_out,
                        const float* __restrict__ b_out,
                        const float* __restrict__ w_res,
                        const float* __restrict__ b_res,
                        float* __restrict__ out)
{
    const int bc   = blockIdx.x;           // 0 .. B*C-1
    const int c    = bc & (C_ - 1);
    const int lane = threadIdx.x & 31;
    const int wave = threadIdx.x >> 5;     // 0..7
    const int h    = lane >> 4;            // half-wave: 0/1
    const int n    = lane & 15;            // N / M row within half
    const int f0   = wave << 4;            // this wave's F-column base

    const float* __restrict__ xs = x + (size_t)bc * (T_ * F_);   // [64][128]
    float* __restrict__ outp     = out + (size_t)bc * (S_ * F_); // [62][128]

    // Per-channel uniform scalars (lower to s_load_*).
    const float wt0 = w_tanh[2 * c + 0], wt1 = w_tanh[2 * c + 1], bt = b_tanh[c];
    const float ws0 = w_sig [2 * c + 0], ws1 = w_sig [2 * c + 1], bs = b_sig [c];
    const float wo  = w_out[c], bo = b_out[c];

    // ---- Load B-fragments: entire 64x16 x sub-tile, K-major fragments.
    // B(4x16) layout: VGPR0 = {K=0 (lanes 0-15), K=2 (lanes 16-31)},
    //                 VGPR1 = {K=1, K=3}; lane%16 = N.
    v2f bfrag[16];
#pragma unroll
    for (int ks = 0; ks < 16; ++ks) {
        const int t = 4 * ks + 2 * h;
        bfrag[ks].x = xs[t * F_ + f0 + n];
        bfrag[ks].y = xs[(t + 1) * F_ + f0 + n];
    }

    // ---- 4 M-tiles of 16 rows; K = 64 in 16 steps of 4.
    // Two accumulators per tile halve the dependent-WMMA chain depth.
#pragma unroll
    for (int m = 0; m < 4; ++m) {
        v8f acc0 = {};
        v8f acc1 = {};
        const int   srow  = 16 * m + n;             // A-matrix row for this lane
        const float amask = (srow < S_) ? 1.0f : 0.0f;
        const int   sclmp = (srow < S_) ? srow : (S_ - 1);
        const float* __restrict__ arow = w_res + sclmp * T_;
#pragma unroll
        for (int ks = 0; ks < 16; ks += 2) {
            const int t0 = 4 * ks + 2 * h;
            const int t1 = t0 + 4;
            // A(16x4) layout: lane%16 = M; VGPR0={K=0,K=2}, VGPR1={K=1,K=3}.
            v2f a0, a1;
            a0.x = arow[t0] * amask;     // branchless S=62 -> 64 zero-pad
            a0.y = arow[t0 + 1] * amask;
            a1.x = arow[t1] * amask;
            a1.y = arow[t1 + 1] * amask;
            acc0 = __builtin_amdgcn_wmma_f32_16x16x4_f32(
                false, a0, false, bfrag[ks],     (short)0, acc0, false, false);
            acc1 = __builtin_amdgcn_wmma_f32_16x16x4_f32(
                false, a1, false, bfrag[ks + 1], (short)0, acc1, false, false);
        }

        // ---- Fused epilogue. C/D layout: VGPR j holds s = 16m + j + 8h.
#pragma unroll
        for (int j = 0; j < 8; ++j) {
            const int s = 16 * m + j + 8 * h;
            if (s < S_) {
                const int fcol = f0 + n;
                const float x0 = xs[s * F_ + fcol];          // WGP$-resident
                const float x1 = xs[(s + D_) * F_ + fcol];
                const float xt = fmaf(x1, wt1, fmaf(x0, wt0, bt));
                const float xv = fmaf(x1, ws1, fmaf(x0, ws0, bs));
                const float g  = fast_tanh(xt) * fast_sigmoid(xv);
                const float r  = (acc0[j] + acc1[j]) + b_res[s];
                __builtin_nontemporal_store(fmaf(g + r, wo, bo),
                                            &outp[s * F_ + fcol]);
            }
        }
    }
}

extern "C" void kernel_launch(void* const* d_in, const int* in_sizes, int n_in,
                              void* d_out, int out_size, void* d_ws, size_t ws_size,
                              hipStream_t stream) {
    (void)in_sizes; (void)n_in; (void)d_ws; (void)ws_size; (void)out_size;
    const float* x      = (const float*)d_in[0];
    const float* w_tanh = (const float*)d_in[1];
    const float* b_tanh = (const float*)d_in[2];
    const float* w_sig  = (const float*)d_in[3];
    const float* b_sig  = (const float*)d_in[4];
    const float* w_out  = (const float*)d_in[5];
    const float* b_out  = (const float*)d_in[6];
    const float* w_res  = (const float*)d_in[7];
    const float* b_res  = (const float*)d_in[8];
    // d_in[9] = dilation (==2), baked in as D_.
    float* out = (float*)d_out;

    dim3 grid(B_ * C_);
    dim3 block(256);
    gated_block_kernel<<<grid, block, 0, stream>>>(
        x, w_tanh, b_tanh, w_sig, b_sig, w_out, b_out, w_res, b_res, out);
}